// GuidedAttention_48395691491480
// MI455X (gfx1250) — compile-verified
//
#include <hip/hip_runtime.h>

// ---------------------------------------------------------------------------
// GuidedAttention on MI455X (gfx1250, wave32).
//
// Roofline: ~0.42 GFLOP vs ~0.22 GB minimum traffic -> pure HBM-bandwidth
// problem (~9.4 us at 23.3 TB/s). f32 end-to-end, so the matrix op is the
// CDNA5-only V_WMMA_F32_16X16X4_F32 for the [1600x128]x[128x8] projection.
//
// The neigh_r output (pure reshape of the 105 MB input) is fused into the
// compute kernel: each 16x128 tile is streamed coalesced once from HBM,
// written out with non-temporal stores, and the WMMA-layout strided reads
// of the same tile then hit L0/L2 instead of HBM. One read + one write of
// neigh total.
// ---------------------------------------------------------------------------

typedef float v2f __attribute__((ext_vector_type(2)));
typedef float v4f __attribute__((ext_vector_type(4)));
typedef float v8f __attribute__((ext_vector_type(8)));

#define BB    128
#define NN    32
#define TT    50
#define HID   128
#define HOPS  8
#define NT    (NN * TT)      // 1600 rows per batch
#define NTILES (NT / 16)     // 100 16-row tiles
#define THREADS 512
#define NWAVES  (THREADS / 32)
#define S1_STRIDE 1601       // odd stride -> conflict-free hop-column scatter

__global__ __launch_bounds__(THREADS)
void guided_attn_kernel(const float* __restrict__ node,   // [B,T,H]
                        const float* __restrict__ neigh,  // [B,N*T,H]
                        const int*   __restrict__ nnum,   // [B]
                        const float* __restrict__ W,      // [HOPS,H]
                        const float* __restrict__ bias,   // [HOPS]
                        float* __restrict__ outBW,        // [B,N*T]
                        float* __restrict__ outBA,        // [B,HOPS,N*T]
                        float* __restrict__ outNR)        // [B,N*T,H]
{
    // Logits / softmax buffer: 8 x 1601 floats = 51,232 B LDS.
    __shared__ float s1_s[HOPS * S1_STRIDE];

    const int b    = blockIdx.x;
    const int tid  = threadIdx.x;
    const int lane = tid & 31;
    const int wave = tid >> 5;
    const int n16  = lane & 15;   // WMMA column (hop index when < 8)
    const int kh   = lane >> 4;   // K-half selector per ISA A/B layout

    const float inv_scale = rsqrtf((float)nnum[0]);

    // ---- Preload B operand (W^T in 4x16 K-chunks) into registers ----------
    // B layout (32-bit 4x16): lanes 0-15 hold N=0..15 with K={0,1} in the two
    // VGPRs; lanes 16-31 hold K={2,3}. Columns 8..15 are zero (HOPS==8).
    v2f Bk[32];
#pragma unroll
    for (int i = 0; i < 32; ++i) {
        v2f bv = {0.0f, 0.0f};
        if (n16 < HOPS) bv = *(const v2f*)&W[n16 * HID + 4 * i + 2 * kh];
        Bk[i] = bv;
    }
    const float biasv = (n16 < HOPS) ? bias[n16] : 0.0f;

    const float* neighb = neigh + (size_t)b * NT * HID;
    const float* nodeb  = node  + (size_t)b * TT * HID;
    float*       nrb    = outNR + (size_t)b * NT * HID;

    // ---- Per tile: stream-copy (reshape output) + WMMA projection ---------
    for (int tile = wave; tile < NTILES; tile += NWAVES) {
        const int r0 = tile * 16;

        // (1) Coalesced stream of the 16x128 tile: neigh -> neigh_r output.
        //     Full 512B rows per wave; NT stores keep the copy out of L2.
        //     This is the single HBM read of neigh; it warms L0/L2 for (2).
        {
            const float* src = neighb + (size_t)r0 * HID;
            float*       dst = nrb    + (size_t)r0 * HID;
#pragma unroll
            for (int r = 0; r < 16; ++r) {
                const v4f* sp = (const v4f*)(src + r * HID) + lane;
                v4f*       dp = (v4f*)(dst + r * HID) + lane;
                __builtin_nontemporal_store(*sp, dp);
            }
        }

        // (2) s1[16x8] = (neigh .* node_tiled)[16x128] x W^T[128x8]
        const int row = r0 + n16;        // A-matrix row this lane supplies
        const int t   = row % TT;        // node_tiled row = nt mod T
        const float* gA = neighb + (size_t)row * HID + 2 * kh;  // L0/L2 hits
        const float* nA = nodeb  + (size_t)t   * HID + 2 * kh;  // 25KB, hot

        v8f c = {};                      // f32 accumulator (8 VGPRs)
#pragma unroll
        for (int i = 0; i < 32; ++i) {   // K = 128 in steps of 4
            v2f g  = *(const v2f*)(gA + 4 * i);
            v2f nd = *(const v2f*)(nA + 4 * i);
            v2f a;
            a.x = g.x * nd.x;            // Hprod on the fly
            a.y = g.y * nd.y;
            // D = A(16x4,f32) x B(4x16,f32) + C  -> v_wmma_f32_16x16x4_f32
            c = __builtin_amdgcn_wmma_f32_16x16x4_f32(
                    false, a, false, Bk[i], (short)0, c, false, false);
        }

        // C layout: VGPR j -> M = j + 8*kh, column = n16. Scatter hop-major
        // logits with bias + 1/sqrt(nn) applied. Stride 1601 -> no conflicts.
        if (n16 < HOPS) {
#pragma unroll
            for (int j = 0; j < 8; ++j) {
                const int m = j + 8 * kh;
                s1_s[n16 * S1_STRIDE + r0 + m] = (c[j] + biasv) * inv_scale;
            }
        }
    }
    __syncthreads();

    // ---- Softmax over N*T per hop: one wave per hop (waves 0..7) ----------
    if (wave < HOPS) {
        float* rowp = &s1_s[wave * S1_STRIDE];

        float mx = -3.402823466e38f;
        for (int j = lane; j < NT; j += 32) mx = fmaxf(mx, rowp[j]);
#pragma unroll
        for (int off = 16; off >= 1; off >>= 1)
            mx = fmaxf(mx, __shfl_xor(mx, off, 32));

        float sum = 0.0f;
        for (int j = lane; j < NT; j += 32) {
            const float e = __expf(rowp[j] - mx);
            rowp[j] = e;
            sum += e;
        }
#pragma unroll
        for (int off = 16; off >= 1; off >>= 1)
            sum += __shfl_xor(sum, off, 32);
        const float inv = 1.0f / sum;

        float* oBA = outBA + ((size_t)b * HOPS + wave) * NT;
        for (int j = lane; j < NT; j += 32) {   // coalesced BA writes
            const float v = rowp[j] * inv;
            rowp[j] = v;
            __builtin_nontemporal_store(v, &oBA[j]);
        }
    }
    __syncthreads();

    // ---- BW = sum over hops of BA ------------------------------------------
    float* oBW = outBW + (size_t)b * NT;
    for (int j = tid; j < NT; j += THREADS) {
        float s = 0.0f;
#pragma unroll
        for (int h = 0; h < HOPS; ++h) s += s1_s[h * S1_STRIDE + j];
        oBW[j] = s;
    }
}

extern "C" void kernel_launch(void* const* d_in, const int* in_sizes, int n_in,
                              void* d_out, int out_size, void* d_ws, size_t ws_size,
                              hipStream_t stream) {
    (void)in_sizes; (void)n_in; (void)out_size; (void)d_ws; (void)ws_size;

    const float* node  = (const float*)d_in[0];  // [B,T,H]      f32
    const float* neigh = (const float*)d_in[1];  // [B,N,T,H]    f32
    const int*   nnum  = (const int*)  d_in[2];  // [B]          i32
    const float* W     = (const float*)d_in[3];  // [HOPS,H]     f32
    const float* bias  = (const float*)d_in[4];  // [HOPS]       f32

    float* out   = (float*)d_out;
    float* outBW = out;                                   // B*N*T
    float* outBA = outBW + (size_t)BB * NT;               // B*HOPS*N*T
    float* outNR = outBA + (size_t)BB * HOPS * NT;        // B*N*T*H

    guided_attn_kernel<<<dim3(BB), dim3(THREADS), 0, stream>>>(
        node, neigh, nnum, W, bias, outBW, outBA, outNR);
}